// CliffordLieIntegrator_87333864997040
// MI455X (gfx1250) — compile-verified
//
#include <hip/hip_runtime.h>
#include <hip/hip_bf16.h>

typedef float v2f __attribute__((ext_vector_type(2)));

#define NTH   288              // threads per block = 9 wave32s
#define TILE  512              // output multivectors per block
#define LSTR  9                // LDS row stride (floats); gcd(9,64)=1 -> conflict-free
#define LROWS (2*NTH + 8)      // 584 rows: 576 computed slots + 8 read-only tail

// ------- Cl(3,0) geometric product on packed pairs, basis [1,e1,e2,e3,e12,e13,e23,e123]
__device__ __forceinline__ void gp8(const v2f* __restrict__ a,
                                    const v2f* __restrict__ b,
                                    v2f* __restrict__ o) {
    o[0] = a[0]*b[0] + a[1]*b[1] + a[2]*b[2] + a[3]*b[3]
         - a[4]*b[4] - a[5]*b[5] - a[6]*b[6] - a[7]*b[7];
    o[1] = a[0]*b[1] + a[1]*b[0] - a[2]*b[4] + a[4]*b[2]
         - a[3]*b[5] + a[5]*b[3] - a[6]*b[7] - a[7]*b[6];
    o[2] = a[0]*b[2] + a[2]*b[0] + a[1]*b[4] - a[4]*b[1]
         - a[3]*b[6] + a[6]*b[3] + a[5]*b[7] + a[7]*b[5];
    o[3] = a[0]*b[3] + a[3]*b[0] + a[1]*b[5] - a[5]*b[1]
         + a[2]*b[6] - a[6]*b[2] - a[4]*b[7] - a[7]*b[4];
    o[4] = a[0]*b[4] + a[4]*b[0] + a[1]*b[2] - a[2]*b[1]
         + a[3]*b[7] + a[7]*b[3] - a[5]*b[6] + a[6]*b[5];
    o[5] = a[0]*b[5] + a[5]*b[0] + a[1]*b[3] - a[3]*b[1]
         - a[2]*b[7] - a[7]*b[2] + a[4]*b[6] - a[6]*b[4];
    o[6] = a[0]*b[6] + a[6]*b[0] + a[2]*b[3] - a[3]*b[2]
         + a[1]*b[7] + a[7]*b[1] - a[4]*b[5] + a[5]*b[4];
    o[7] = a[0]*b[7] + a[7]*b[0] + a[1]*b[6] + a[6]*b[1]
         - a[2]*b[5] - a[5]*b[2] + a[3]*b[4] + a[4]*b[3];
}

// exp(u) = 1 + sum_{n=1..7} u^n / n!
__device__ __forceinline__ void cexp8(const v2f* __restrict__ u, v2f* __restrict__ r) {
    v2f term[8], t[8];
#pragma unroll
    for (int c = 0; c < 8; ++c) { term[c] = u[c]; r[c] = u[c]; }   // n=1 term
    r[0] += 1.0f;                                                   // identity
#pragma unroll
    for (int n = 2; n < 8; ++n) {
        gp8(term, u, t);
        const float inv = 1.0f / (float)n;
#pragma unroll
        for (int c = 0; c < 8; ++c) { term[c] = t[c] * inv; r[c] += term[c]; }
    }
}

// dexp^{-1}_u(w) = w - 1/2 ad + 1/12 ad^2 + 0*ad^3 - 1/720 ad^4
__device__ __forceinline__ void dexpinv8(const v2f* __restrict__ u,
                                         const v2f* __restrict__ w,
                                         v2f* __restrict__ r) {
    v2f ad[8], t1[8], t2[8];
#pragma unroll
    for (int c = 0; c < 8; ++c) { ad[c] = w[c]; r[c] = w[c]; }
    gp8(u, ad, t1); gp8(ad, u, t2);
#pragma unroll
    for (int c = 0; c < 8; ++c) { ad[c] = t1[c] - t2[c]; r[c] += -0.5f * ad[c]; }
    gp8(u, ad, t1); gp8(ad, u, t2);
#pragma unroll
    for (int c = 0; c < 8; ++c) { ad[c] = t1[c] - t2[c]; r[c] += (1.0f/12.0f) * ad[c]; }
    gp8(u, ad, t1); gp8(ad, u, t2);
#pragma unroll
    for (int c = 0; c < 8; ++c) { ad[c] = t1[c] - t2[c]; }          // B3 = 0
    gp8(u, ad, t1); gp8(ad, u, t2);
#pragma unroll
    for (int c = 0; c < 8; ++c) { ad[c] = t1[c] - t2[c]; r[c] += (-1.0f/720.0f) * ad[c]; }
}

// vector_field for the pair (pA, pB): derivative(z) + 0.05/4 * sum_s gp(z_p, z_{p+s})/s
__device__ __forceinline__ void vfield8(const v2f* __restrict__ zo,
                                        const float* __restrict__ zb,
                                        int pA, int pB, v2f* __restrict__ w) {
    v2f acc[8], nb[8], t[8];
#pragma unroll
    for (int c = 0; c < 8; ++c) acc[c] = (v2f)0.0f;
#pragma unroll
    for (int si = 0; si < 4; ++si) {
        const int s = 1 << si;
        const float* nA = zb + (pA + s) * LSTR;
        const float* nB = zb + (pB + s) * LSTR;
#pragma unroll
        for (int c = 0; c < 8; ++c) { v2f v; v.x = nA[c]; v.y = nB[c]; nb[c] = v; }
        gp8(zo, nb, t);
        const float inv = 1.0f / (float)s;      // exact powers of two
#pragma unroll
        for (int c = 0; c < 8; ++c) acc[c] += t[c] * inv;
    }
    const float IS = 0.05f / 4.0f;
    w[0] =  IS * acc[0];
    w[1] =  zo[4] + IS * acc[1];
    w[2] =  zo[5] + IS * acc[2];
    w[3] =  zo[6] + IS * acc[3];
    w[4] = -zo[1] + IS * acc[4];
    w[5] = -zo[2] + IS * acc[5];
    w[6] = -zo[3] + IS * acc[6];
    w[7] =  IS * acc[7];
}

// CDNA5 async global->LDS DMA of one multivector row (8 dwords, tracked by ASYNCcnt)
__device__ __forceinline__ void async_row(unsigned lds, const float* g) {
    asm volatile(
        "global_load_async_to_lds_b32 %0, %1, off\n\t"
        "global_load_async_to_lds_b32 %0, %1, off offset:4\n\t"
        "global_load_async_to_lds_b32 %0, %1, off offset:8\n\t"
        "global_load_async_to_lds_b32 %0, %1, off offset:12\n\t"
        "global_load_async_to_lds_b32 %0, %1, off offset:16\n\t"
        "global_load_async_to_lds_b32 %0, %1, off offset:20\n\t"
        "global_load_async_to_lds_b32 %0, %1, off offset:24\n\t"
        "global_load_async_to_lds_b32 %0, %1, off offset:28"
        :: "v"(lds), "v"(g) : "memory");
}

__device__ __forceinline__ void store_z(float* __restrict__ zb, int pA, int pB,
                                        const v2f* __restrict__ zo) {
#pragma unroll
    for (int c = 0; c < 8; ++c) {
        zb[pA * LSTR + c] = zo[c].x;
        zb[pB * LSTR + c] = zo[c].y;
    }
}

__global__ __launch_bounds__(NTH)
void clifford_rkmk4_pk_kernel(const float* __restrict__ yin, float* __restrict__ yout) {
    __shared__ float zb[LROWS * LSTR];

    const int t     = threadIdx.x;
    const int row   = blockIdx.y;
    const int tile0 = blockIdx.x * TILE;
    const int pA = t;
    const int pB = t + NTH;
    const int elemA = (tile0 + pA) & 65535;      // jnp.roll wraparound (axis len 65536)
    const int elemB = (tile0 + pB) & 65535;
    const size_t rbase = ((size_t)row << 19);    // 65536*8

    async_row((unsigned)(size_t)(const void*)&zb[pA * LSTR], yin + rbase + (size_t)elemA * 8);
    async_row((unsigned)(size_t)(const void*)&zb[pB * LSTR], yin + rbase + (size_t)elemB * 8);
    asm volatile("s_wait_asynccnt 0" ::: "memory");
    __syncthreads();

    v2f y[8];
#pragma unroll
    for (int c = 0; c < 8; ++c) { v2f v; v.x = zb[pA*LSTR+c]; v.y = zb[pB*LSTR+c]; y[c] = v; }

    const float dt = 0.1f;
    v2f k[8], u[8], acc[8], zo[8], w[8], e[8];

    // ---------------- stage 1: u=0 -> z1 = y, k1 = vf(y) ----------------
    vfield8(y, zb, pA, pB, w);
#pragma unroll
    for (int c = 0; c < 8; ++c) { k[c] = w[c]; u[c] = (0.5f * dt) * k[c]; acc[c] = k[c]; }
    __syncthreads();                    // all reads of z1(=y) complete before overwrite
    cexp8(u, e); gp8(e, y, zo);         // z2 = exp(u2) * y
    store_z(zb, pA, pB, zo);
    __syncthreads();

    // ---------------- stage 2 ----------------
    vfield8(zo, zb, pA, pB, w);
    dexpinv8(u, w, k);
#pragma unroll
    for (int c = 0; c < 8; ++c) { u[c] = (0.5f * dt) * k[c]; acc[c] += 2.0f * k[c]; }
    __syncthreads();
    cexp8(u, e); gp8(e, y, zo);         // z3 = exp(u3) * y
    store_z(zb, pA, pB, zo);
    __syncthreads();

    // ---------------- stage 3 ----------------
    vfield8(zo, zb, pA, pB, w);
    dexpinv8(u, w, k);
#pragma unroll
    for (int c = 0; c < 8; ++c) { u[c] = dt * k[c]; acc[c] += 2.0f * k[c]; }
    __syncthreads();
    cexp8(u, e); gp8(e, y, zo);         // z4 = exp(u4) * y
    store_z(zb, pA, pB, zo);
    __syncthreads();

    // ---------------- stage 4 + combine + output ----------------
    vfield8(zo, zb, pA, pB, w);
    dexpinv8(u, w, k);
#pragma unroll
    for (int c = 0; c < 8; ++c) {
        acc[c] += k[c];
        v2f x = (dt * acc[c]) / 6.0f;
        float xa = x.x, xb = x.y;
        xa = (xa != xa) ? 0.0f : xa;  xa = fminf(1.0f, fmaxf(-1.0f, xa));
        xb = (xb != xb) ? 0.0f : xb;  xb = fminf(1.0f, fmaxf(-1.0f, xb));
        x.x = xa; x.y = xb; u[c] = x;
    }
    cexp8(u, e);
    v2f o[8];
    gp8(e, y, o);

    float* dstA = yout + rbase + (size_t)elemA * 8;
    *(float4*)(dstA)     = make_float4(o[0].x, o[1].x, o[2].x, o[3].x);
    *(float4*)(dstA + 4) = make_float4(o[4].x, o[5].x, o[6].x, o[7].x);
    if (pB < TILE) {                                   // B half: only first 224 threads output
        float* dstB = yout + rbase + (size_t)elemB * 8;
        *(float4*)(dstB)     = make_float4(o[0].y, o[1].y, o[2].y, o[3].y);
        *(float4*)(dstB + 4) = make_float4(o[4].y, o[5].y, o[6].y, o[7].y);
    }
}

extern "C" void kernel_launch(void* const* d_in, const int* in_sizes, int n_in,
                              void* d_out, int out_size, void* d_ws, size_t ws_size,
                              hipStream_t stream) {
    const float* y = (const float*)d_in[0];
    float* out = (float*)d_out;
    const int rows = in_sizes[0] >> 19;          // elements / (65536*8)
    dim3 grid(65536 / TILE, rows, 1);
    dim3 block(NTH, 1, 1);
    clifford_rkmk4_pk_kernel<<<grid, block, 0, stream>>>(y, out);
}